// Attention_85289460564564
// MI455X (gfx1250) — compile-verified
//
#include <hip/hip_runtime.h>
#include <cstddef>

// ---- CDNA5 f32 WMMA (V_WMMA_F32_16X16X4_F32), wave32 ----
typedef float v2f __attribute__((ext_vector_type(2)));
typedef float v8f __attribute__((ext_vector_type(8)));

#define NEG_MAX -3.402823466e38f

static __device__ __forceinline__ v8f wmma_f32(v2f a, v2f b, v8f c) {
  // 8 args: (neg_a, A, neg_b, B, c_mod, C, reuse_a, reuse_b)
  return __builtin_amdgcn_wmma_f32_16x16x4_f32(false, a, false, b, (short)0, c,
                                               false, false);
}

// Scheduler grouping hints: batch VMEM reads ahead of WMMA bursts so the
// backend doesn't serialize load -> s_wait_loadcnt 0 -> wmma per fragment.
// mask 0x20 = VMEM read, 0x8 = MFMA/WMMA. No-op if builtin unavailable.
#if defined(__has_builtin)
#if __has_builtin(__builtin_amdgcn_sched_group_barrier)
#define SCHED_LOADS(n) __builtin_amdgcn_sched_group_barrier(0x020, (n), 0)
#define SCHED_WMMAS(n) __builtin_amdgcn_sched_group_barrier(0x008, (n), 0)
#endif
#endif
#ifndef SCHED_LOADS
#define SCHED_LOADS(n)
#define SCHED_WMMAS(n)
#endif

// Shapes (hardcoded from reference): B=2 N=2048 DIM=1024 H=16 DH=64 M=16 J=2064
// Layout conventions:
//   q_lin/k_lin/v_lin/oh : (b, n, h, d) contiguous, i.e. (4096, 1024) row-major
//   pre/post (attention planes): (b, h, i, j) with j stride 1, J = 2064
//
// f32 WMMA register mapping (wave32):
//   A (16x4):  lane L, reg r -> A[L&15][(L>>4)*2 + r]
//   B (4x16):  lane L, reg r -> B[(L>>4)*2 + r][L&15]
//   C/D(16x16):lane L, reg r -> D[(L>>4)*8 + r][L&15]

// ---------------------------------------------------------------------------
// Generic C = A(rows,K) * Bw(cols,K)^T [+ bias], one wave per 16x64 tile.
// Software-pipelined: "next" k-step fragments loaded into distinct registers
// before the "current" WMMAs are issued, so loads overlap matrix ops.
// ---------------------------------------------------------------------------
__global__ __launch_bounds__(32) void gemm16x64_nt(
    const float* __restrict__ A, const float* __restrict__ Bw,
    const float* __restrict__ bias, float* __restrict__ C, int K, int ldc) {
  const int lane = threadIdx.x & 31;
  const int m = lane & 15;
  const int hi = lane >> 4;
  const int i0 = blockIdx.x * 16;
  const int o0 = blockIdx.y * 64;
  const float* arow = A + (size_t)(i0 + m) * K + hi * 2;
  const float* b0p = Bw + (size_t)(o0 + m) * K + hi * 2;
  const float* b1p = Bw + (size_t)(o0 + 16 + m) * K + hi * 2;
  const float* b2p = Bw + (size_t)(o0 + 32 + m) * K + hi * 2;
  const float* b3p = Bw + (size_t)(o0 + 48 + m) * K + hi * 2;

  v2f a_c  = *(const v2f*)(arow);
  v2f b0_c = *(const v2f*)(b0p);
  v2f b1_c = *(const v2f*)(b1p);
  v2f b2_c = *(const v2f*)(b2p);
  v2f b3_c = *(const v2f*)(b3p);
  v8f acc0 = {}, acc1 = {}, acc2 = {}, acc3 = {};

  for (int kb = 0; kb < K - 4; kb += 4) {
    const int kn = kb + 4;
    v2f a_n  = *(const v2f*)(arow + kn);
    v2f b0_n = *(const v2f*)(b0p + kn);
    v2f b1_n = *(const v2f*)(b1p + kn);
    v2f b2_n = *(const v2f*)(b2p + kn);
    v2f b3_n = *(const v2f*)(b3p + kn);
    acc0 = wmma_f32(a_c, b0_c, acc0);
    acc1 = wmma_f32(a_c, b1_c, acc1);
    acc2 = wmma_f32(a_c, b2_c, acc2);
    acc3 = wmma_f32(a_c, b3_c, acc3);
    SCHED_LOADS(5);  // all 5 next-step loads first ...
    SCHED_WMMAS(4);  // ... then the 4 current-step WMMAs
    a_c = a_n; b0_c = b0_n; b1_c = b1_n; b2_c = b2_n; b3_c = b3_n;
  }
  acc0 = wmma_f32(a_c, b0_c, acc0);
  acc1 = wmma_f32(a_c, b1_c, acc1);
  acc2 = wmma_f32(a_c, b2_c, acc2);
  acc3 = wmma_f32(a_c, b3_c, acc3);

  const float bv0 = bias ? bias[o0 + m] : 0.f;
  const float bv1 = bias ? bias[o0 + 16 + m] : 0.f;
  const float bv2 = bias ? bias[o0 + 32 + m] : 0.f;
  const float bv3 = bias ? bias[o0 + 48 + m] : 0.f;
#pragma unroll
  for (int r = 0; r < 8; ++r) {
    float* crow = C + (size_t)(i0 + hi * 8 + r) * ldc;
    crow[o0 + m] = acc0[r] + bv0;
    crow[o0 + 16 + m] = acc1[r] + bv1;
    crow[o0 + 32 + m] = acc2[r] + bv2;
    crow[o0 + 48 + m] = acc3[r] + bv3;
  }
}

// ---------------------------------------------------------------------------
// Rotary: in-place on first 32 dims of each head of q_lin and k_lin.
// 1,048,576 threads; t = (((b*2048+n)*16+h)*16+d)
// ---------------------------------------------------------------------------
__global__ __launch_bounds__(256) void rotary_kernel(float* __restrict__ q,
                                                     float* __restrict__ k,
                                                     const float* __restrict__ rp) {
  const int t = blockIdx.x * 256 + threadIdx.x;
  const int d = t & 15;
  const int h = (t >> 4) & 15;
  const int n = (t >> 8) & 2047;
  const int b = t >> 19;
  const float fr = rp[n * 32 + d];
  const float c = cosf(fr), s = sinf(fr);
  const size_t off = (size_t)(b * 2048 + n) * 1024 + h * 64 + d;
  float q0 = q[off], q1 = q[off + 16];
  q[off] = q0 * c - q1 * s;
  q[off + 16] = q1 * c + q0 * s;
  float k0 = k[off], k1 = k[off + 16];
  k[off] = k0 * c - k1 * s;
  k[off + 16] = k1 * c + k0 * s;
}

// ---------------------------------------------------------------------------
// dots[b,h,i,j] = scale * sum_d q[b,h,i,d]*kcat[b,h,j,d] -> pre_softmax_attn
// kcat: j<16 -> mem_k, else k_lin.  grid (129, 128, B*H), one wave per tile.
// Double-buffered over the 16 k-steps (DH=64).
// ---------------------------------------------------------------------------
__global__ __launch_bounds__(32) void dots_kernel(const float* __restrict__ q_lin,
                                                  const float* __restrict__ k_lin,
                                                  const float* __restrict__ mem_k,
                                                  float* __restrict__ pre) {
  const int lane = threadIdx.x & 31;
  const int m = lane & 15;
  const int hi = lane >> 4;
  const int j0 = blockIdx.x * 16;
  const int i0 = blockIdx.y * 16;
  const int b = blockIdx.z >> 4;
  const int h = blockIdx.z & 15;
  const float* arow = q_lin + (size_t)(b * 2048 + i0 + m) * 1024 + h * 64 + hi * 2;
  const float* brow =
      ((j0 == 0) ? (mem_k + (size_t)(h * 16 + m) * 64)
                 : (k_lin + (size_t)(b * 2048 + j0 - 16 + m) * 1024 + h * 64)) +
      hi * 2;
  v2f a_c = *(const v2f*)(arow);
  v2f b_c = *(const v2f*)(brow);
  v8f acc = {};
#pragma unroll
  for (int kb = 0; kb < 60; kb += 4) {
    v2f a_n = *(const v2f*)(arow + kb + 4);
    v2f b_n = *(const v2f*)(brow + kb + 4);
    acc = wmma_f32(a_c, b_c, acc);
    SCHED_LOADS(2);
    SCHED_WMMAS(1);
    a_c = a_n;
    b_c = b_n;
  }
  acc = wmma_f32(a_c, b_c, acc);

  const float scale = 0.125f;  // 64^-0.5
#pragma unroll
  for (int r = 0; r < 8; ++r) {
    pre[((size_t)(b * 16 + h) * 2048 + i0 + hi * 8 + r) * 2064 + j0 + m] =
        acc[r] * scale;
  }
}

// ---------------------------------------------------------------------------
// Head mix over h via WMMA (A = pre_proj^T, K = 16 heads) + causal mask.
// mixed[b,kk,i,j] = sum_h preP[h,kk]*pre[b,h,i,j]; mask where j > i+16.
// grid (129, 2048, 2), one wave per (i, 16-j tile); writes into `post` (temp).
// All loads gathered up front, then one WMMA burst.
// ---------------------------------------------------------------------------
__global__ __launch_bounds__(32) void premix_kernel(const float* __restrict__ pre,
                                                    const float* __restrict__ preP,
                                                    float* __restrict__ post) {
  const int lane = threadIdx.x & 31;
  const int m = lane & 15;
  const int hi = lane >> 4;
  const int j0 = blockIdx.x * 16;
  const int i = blockIdx.y;
  const int b = blockIdx.z;
  v2f a[4], bv[4];
#pragma unroll
  for (int t = 0; t < 4; ++t) {
    const int kh = 4 * t + hi * 2;
    a[t].x = preP[kh * 16 + m];  // pre_proj[h=kh][kk=m]
    a[t].y = preP[(kh + 1) * 16 + m];
    bv[t].x = pre[((size_t)(b * 16 + kh) * 2048 + i) * 2064 + j0 + m];
    bv[t].y = pre[((size_t)(b * 16 + kh + 1) * 2048 + i) * 2064 + j0 + m];
  }
  v8f acc = {};
#pragma unroll
  for (int t = 0; t < 4; ++t) acc = wmma_f32(a[t], bv[t], acc);

  const int j = j0 + m;
  const bool msk = (j > i + 16);
#pragma unroll
  for (int r = 0; r < 8; ++r) {
    post[((size_t)(b * 16 + hi * 8 + r) * 2048 + i) * 2064 + j] =
        msk ? NEG_MAX : acc[r];
  }
}

// ---------------------------------------------------------------------------
// Row softmax over j (2064) in-place on `post`. One 256-thread block per row.
// ---------------------------------------------------------------------------
__global__ __launch_bounds__(256) void softmax_kernel(float* __restrict__ post) {
  __shared__ float buf[2064];
  __shared__ float red[256];
  const int tid = threadIdx.x;
  const size_t base = (size_t)blockIdx.x * 2064;
  for (int j = tid; j < 2064; j += 256) buf[j] = post[base + j];
  __syncthreads();
  float lm = NEG_MAX;
  for (int j = tid; j < 2064; j += 256) lm = fmaxf(lm, buf[j]);
  red[tid] = lm;
  __syncthreads();
  for (int s = 128; s > 0; s >>= 1) {
    if (tid < s) red[tid] = fmaxf(red[tid], red[tid + s]);
    __syncthreads();
  }
  const float mx = red[0];
  __syncthreads();
  float ls = 0.f;
  for (int j = tid; j < 2064; j += 256) {
    float e = __expf(buf[j] - mx);
    buf[j] = e;
    ls += e;
  }
  red[tid] = ls;
  __syncthreads();
  for (int s = 128; s > 0; s >>= 1) {
    if (tid < s) red[tid] += red[tid + s];
    __syncthreads();
  }
  const float inv = 1.0f / red[0];
  __syncthreads();
  for (int j = tid; j < 2064; j += 256) post[base + j] = buf[j] * inv;
}

// ---------------------------------------------------------------------------
// Fused post-mix + attn@v. Workgroup = (b, 16-row i tile), 16 waves.
// Per 16-j chunk: wave w WMMA-mixes attn2[kk][ii=w][jj] via post_proj^T,
// stages it in LDS, barrier, then wave w (= head kk) accumulates its
// 16x64 output tile attn2 x v with 16 WMMAs. Reads `post` exactly once.
// Loads are hoisted ahead of each WMMA burst so they can overlap.
// ---------------------------------------------------------------------------
__global__ __launch_bounds__(512) void av_kernel(const float* __restrict__ post,
                                                 const float* __restrict__ postP,
                                                 const float* __restrict__ v_lin,
                                                 const float* __restrict__ mem_v,
                                                 float* __restrict__ oh) {
  __shared__ float a2[16 * 16 * 16];  // [kk][ii][jj]
  const int w = threadIdx.x >> 5;     // wave id / head
  const int lane = threadIdx.x & 31;
  const int m = lane & 15;
  const int hi = lane >> 4;
  const int i0 = blockIdx.x * 16;
  const int b = blockIdx.y;
  v2f am[4];
#pragma unroll
  for (int t = 0; t < 4; ++t) {
    const int kh = 4 * t + hi * 2;
    am[t].x = postP[kh * 16 + m];  // post_proj[h=kh][kk=m]
    am[t].y = postP[(kh + 1) * 16 + m];
  }
  v8f acc0 = {}, acc1 = {}, acc2 = {}, acc3 = {};
  for (int j0 = 0; j0 < 2064; j0 += 16) {
    // --- stage: attn2 for row ii = w, all 16 kk, 16 j ---
    v2f bv[4];
#pragma unroll
    for (int t = 0; t < 4; ++t) {
      const int kh = 4 * t + hi * 2;
      bv[t].x = post[((size_t)(b * 16 + kh) * 2048 + i0 + w) * 2064 + j0 + m];
      bv[t].y = post[((size_t)(b * 16 + kh + 1) * 2048 + i0 + w) * 2064 + j0 + m];
    }
    v8f mix = {};
#pragma unroll
    for (int t = 0; t < 4; ++t) mix = wmma_f32(am[t], bv[t], mix);
    SCHED_LOADS(8);
    SCHED_WMMAS(4);
#pragma unroll
    for (int r = 0; r < 8; ++r) a2[(hi * 8 + r) * 256 + w * 16 + m] = mix[r];
    __syncthreads();
    // --- GEMM: out[kk=w, 16i, 64d] += attn2[16i,16j] * v[16j,64d] ---
    v2f aj[4], bb[4][4];
#pragma unroll
    for (int t = 0; t < 4; ++t) {
      const int kj = 4 * t + hi * 2;
      const int j = j0 + kj;
      const float* vr0;
      const float* vr1;
      if (j0 == 0) {  // whole chunk is the memory slots (M=16)
        vr0 = mem_v + (size_t)(w * 16 + j) * 64;
        vr1 = vr0 + 64;
      } else {
        vr0 = v_lin + (size_t)(b * 2048 + j - 16) * 1024 + w * 64;
        vr1 = vr0 + 1024;
      }
      bb[t][0].x = vr0[m];      bb[t][0].y = vr1[m];
      bb[t][1].x = vr0[16 + m]; bb[t][1].y = vr1[16 + m];
      bb[t][2].x = vr0[32 + m]; bb[t][2].y = vr1[32 + m];
      bb[t][3].x = vr0[48 + m]; bb[t][3].y = vr1[48 + m];
      aj[t] = *(const v2f*)&a2[w * 256 + m * 16 + kj];
    }
#pragma unroll
    for (int t = 0; t < 4; ++t) {
      acc0 = wmma_f32(aj[t], bb[t][0], acc0);
      acc1 = wmma_f32(aj[t], bb[t][1], acc1);
      acc2 = wmma_f32(aj[t], bb[t][2], acc2);
      acc3 = wmma_f32(aj[t], bb[t][3], acc3);
    }
    SCHED_LOADS(32);
    SCHED_WMMAS(16);
    __syncthreads();
  }
#pragma unroll
  for (int r = 0; r < 8; ++r) {
    float* orow = oh + (size_t)(b * 2048 + i0 + hi * 8 + r) * 1024 + w * 64;
    orow[m] = acc0[r];
    orow[16 + m] = acc1[r];
    orow[32 + m] = acc2[r];
    orow[48 + m] = acc3[r];
  }
}

// ---------------------------------------------------------------------------
extern "C" void kernel_launch(void* const* d_in, const int* in_sizes, int n_in,
                              void* d_out, int out_size, void* d_ws,
                              size_t ws_size, hipStream_t stream) {
  (void)in_sizes; (void)n_in; (void)out_size; (void)ws_size;
  const float* x     = (const float*)d_in[0];
  const float* rp    = (const float*)d_in[1];
  const float* Wq    = (const float*)d_in[2];
  const float* Wk    = (const float*)d_in[3];
  const float* Wv    = (const float*)d_in[4];
  const float* mem_k = (const float*)d_in[5];
  const float* mem_v = (const float*)d_in[6];
  const float* preP  = (const float*)d_in[7];
  const float* postP = (const float*)d_in[8];
  const float* Wo    = (const float*)d_in[9];
  const float* bo    = (const float*)d_in[10];

  float* out  = (float*)d_out;    // (2,2048,1024)
  float* pre  = out + 4194304;    // (2,16,2048,2064)
  float* post = pre + 135266304;  // (2,16,2048,2064)

  float* ws    = (float*)d_ws;    // 64 MB used
  float* q_lin = ws;              // (4096,1024)
  float* k_lin = ws + 4194304;
  float* v_lin = ws + 8388608;
  float* oh    = ws + 12582912;

  const dim3 gP(256, 16);  // 4096/16 row tiles x 1024/64 col tiles
  gemm16x64_nt<<<gP, 32, 0, stream>>>(x, Wq, nullptr, q_lin, 1024, 1024);
  gemm16x64_nt<<<gP, 32, 0, stream>>>(x, Wk, nullptr, k_lin, 1024, 1024);
  gemm16x64_nt<<<gP, 32, 0, stream>>>(x, Wv, nullptr, v_lin, 1024, 1024);

  rotary_kernel<<<4096, 256, 0, stream>>>(q_lin, k_lin, rp);

  dots_kernel<<<dim3(129, 128, 32), 32, 0, stream>>>(q_lin, k_lin, mem_k, pre);

  premix_kernel<<<dim3(129, 2048, 2), 32, 0, stream>>>(pre, preP, post);

  softmax_kernel<<<65536, 256, 0, stream>>>(post);

  av_kernel<<<dim3(128, 2), 512, 0, stream>>>(post, postP, v_lin, mem_v, oh);

  gemm16x64_nt<<<gP, 32, 0, stream>>>(oh, Wo, bo, out, 1024, 1024);
}